// ReactionForge_5188320494204
// MI455X (gfx1250) — compile-verified
//
#include <hip/hip_runtime.h>
#include <hip/hip_bf16.h>
#include <math.h>

typedef __attribute__((ext_vector_type(2))) float v2f;
typedef __attribute__((ext_vector_type(8))) float v8f;

#define NN 150000
#define EE 600000
#define BB 4096
#define CC 128
#define NTY 4
#define NLAY 3
#define KMAX 384    // largest K used by the WMMA GEMM (fusion / GRU input)

// ---------------------------------------------------------------------------
// Workgroup WMMA GEMM: C[M x Nc] = act(A[M x K] @ B + bias)
//   - 8 waves / block; block handles a 16-row stripe x up-to-128 columns
//   - A tile staged in LDS once, shared by all 8 waves (8x less HBM traffic)
//   - B is [K x Nc] row-major (TRANSB=0) or [Nc x K] row-major (TRANSB=1)
//   - branch-free K-loop of V_WMMA_F32_16X16X4_F32
// ---------------------------------------------------------------------------
template<int TRANSB, int ACT>
__global__ __launch_bounds__(256)
void gemm_wmma_wg(const float* __restrict__ A, const float* __restrict__ Bm,
                  const float* __restrict__ bias, float* __restrict__ C,
                  int M, int Nc, int K)
{
    __shared__ float aS[16][KMAX + 4];   // stride mod 64 = 4 words -> conflict-free

    const int tid = threadIdx.x;
    const int tm  = blockIdx.x;

    // cooperative stage of the 16 x K activation tile
    for (int r = 0; r < 16; ++r) {
        int grow = tm * 16 + r;
        if (grow >= M) grow = M - 1;
        for (int c = tid; c < K; c += 256)
            aS[r][c] = A[(size_t)grow * K + c];
    }
    __syncthreads();

    const int w    = tid >> 5;                    // wave id 0..7
    const int lane = tid & 31;
    const int half = lane >> 4;
    const int l16  = lane & 15;
    const int ntile = blockIdx.y * 8 + w;         // 16-col tile index
    if (ntile * 16 >= Nc) return;                 // whole-wave uniform exit
    const int col = ntile * 16 + l16;

    v8f acc = {};
    for (int k0 = 0; k0 < K; k0 += 4) {
        const int ka = k0 + 2 * half;             // lanes0-15: K=k0,k0+1 ; 16-31: +2,+3
        v2f a, b;
        a.x = aS[l16][ka];
        a.y = aS[l16][ka + 1];
        if (TRANSB) {
            b.x = Bm[(size_t)col * K + ka];
            b.y = Bm[(size_t)col * K + ka + 1];
        } else {
            b.x = Bm[(size_t)ka * Nc + col];
            b.y = Bm[(size_t)(ka + 1) * Nc + col];
        }
        acc = __builtin_amdgcn_wmma_f32_16x16x4_f32(false, a, false, b,
                                                    (short)0, acc, false, false);
    }
    const float bv = bias[col];
    #pragma unroll
    for (int r = 0; r < 8; ++r) {
        int orow = tm * 16 + r + 8 * half;        // C: VGPR r -> M=r / r+8
        if (orow < M) {
            float v = acc[r] + bv;
            if (ACT) v = fmaxf(v, 0.0f);
            C[(size_t)orow * Nc + col] = v;
        }
    }
}

static inline void launch_gemm(const float* A, const float* Bm, const float* bias,
                               float* C, int M, int Nc, int K, int transB, int act,
                               hipStream_t s)
{
    dim3 grid((M + 15) / 16, (Nc + 127) / 128);
    if (transB)      gemm_wmma_wg<1, 0><<<grid, 256, 0, s>>>(A, Bm, bias, C, M, Nc, K);
    else if (act)    gemm_wmma_wg<0, 1><<<grid, 256, 0, s>>>(A, Bm, bias, C, M, Nc, K);
    else             gemm_wmma_wg<0, 0><<<grid, 256, 0, s>>>(A, Bm, bias, C, M, Nc, K);
}

// ---------------------------------------------------------------------------
// Edge bucketing by type (types constant across layers -> done once per graph)
// int scratch layout: [0..3]=cnt, [4..7]=cursor, [8..12]=off, [13..17]=tileOff
// ---------------------------------------------------------------------------
__global__ void bucket_init(int* ints) { if (threadIdx.x < 20) ints[threadIdx.x] = 0; }

__global__ void bucket_count(const int* __restrict__ ea, int E, int* ints) {
    int e = blockIdx.x * 256 + threadIdx.x;
    if (e < E) atomicAdd(&ints[ea[e]], 1);
}

__global__ void bucket_offsets(int* ints) {
    if (blockIdx.x == 0 && threadIdx.x == 0) {
        int* cnt = ints; int* off = ints + 8; int* toff = ints + 13;
        off[0] = 0; toff[0] = 0;
        for (int t = 0; t < NTY; ++t) {
            off[t + 1]  = off[t] + cnt[t];
            toff[t + 1] = toff[t] + (cnt[t] + 15) / 16;
        }
    }
}

__global__ void bucket_scatter(const int* __restrict__ ea, int E, int* ints,
                               int* __restrict__ ids) {
    int e = blockIdx.x * 256 + threadIdx.x;
    if (e < E) {
        int t = ea[e];
        int pos = ints[8 + t] + atomicAdd(&ints[4 + t], 1);
        ids[pos] = e;
    }
}

// ---------------------------------------------------------------------------
// Edge message MLP (WMMA double-GEMM through LDS) + scatter-add by col.
//   cat = [h[row] || h[col]] (16x256)  ->  h1 = relu(cat@W1_t + b1_t) (16x128)
//   m = h1@W2_t + b2_t (16x128)        ->  atomicAdd into out[col]
// One workgroup (8 waves) per tile of 16 same-type edges.
// ---------------------------------------------------------------------------
__global__ __launch_bounds__(256)
void edge_mlp_wmma(const float* __restrict__ h,
                   const int* __restrict__ row_idx, const int* __restrict__ col_idx,
                   const int* __restrict__ ids, const int* __restrict__ meta,
                   const float* __restrict__ W1, const float* __restrict__ b1,
                   const float* __restrict__ W2, const float* __restrict__ b2,
                   float* __restrict__ out)
{
    const int* tileOff = meta + 13;
    int bid = blockIdx.x;
    if (bid >= tileOff[NTY]) return;            // uniform exit, EXEC stays full below
    int t = 0;
    while (bid >= tileOff[t + 1]) ++t;
    const int* off = meta + 8;
    int estart = off[t] + (bid - tileOff[t]) * 16;
    int ecount = off[t + 1] - estart; if (ecount > 16) ecount = 16;

    const float* W1t = W1 + (size_t)t * 256 * CC;
    const float* b1t = b1 + (size_t)t * CC;
    const float* W2t = W2 + (size_t)t * CC * CC;
    const float* b2t = b2 + (size_t)t * CC;

    __shared__ float catS[16][260];   // 16x256 (+pad, stride%64=4 -> conflict-free)
    __shared__ float h1S[16][132];    // 16x128 (+pad)
    __shared__ int   rS[16], cS[16];

    int tid = threadIdx.x;
    if (tid < 16) {
        int ii = tid < ecount ? tid : (ecount - 1);
        int eid = ids[estart + ii];
        rS[tid] = row_idx[eid];
        cS[tid] = col_idx[eid];
    }
    __syncthreads();

    // gather concat tile
    for (int idx = tid; idx < 16 * 256; idx += 256) {
        int r = idx >> 8, c = idx & 255;
        catS[r][c] = (c < CC) ? h[(size_t)rS[r] * CC + c]
                              : h[(size_t)cS[r] * CC + (c - CC)];
    }
    __syncthreads();

    int w    = tid >> 5;        // wave id 0..7 -> output cols [16w,16w+16)
    int lane = tid & 31;
    int half = lane >> 4;
    int l16  = lane & 15;
    int ncol = w * 16 + l16;

    // GEMM1: 16x256 @ 256x128 (this wave's 16-col slice), fused bias+relu
    v8f acc = {};
    for (int k0 = 0; k0 < 256; k0 += 4) {
        int ka = k0 + 2 * half;
        v2f a, b;
        a.x = catS[l16][ka];
        a.y = catS[l16][ka + 1];
        b.x = W1t[(size_t)ka * CC + ncol];
        b.y = W1t[(size_t)(ka + 1) * CC + ncol];
        acc = __builtin_amdgcn_wmma_f32_16x16x4_f32(false, a, false, b,
                                                    (short)0, acc, false, false);
    }
    float bb1 = b1t[ncol];
    #pragma unroll
    for (int r = 0; r < 8; ++r) {
        int rr = r + 8 * half;
        h1S[rr][ncol] = fmaxf(acc[r] + bb1, 0.0f);
    }
    __syncthreads();

    // GEMM2: 16x128 @ 128x128 slice, then scatter-add
    v8f acc2 = {};
    for (int k0 = 0; k0 < CC; k0 += 4) {
        int ka = k0 + 2 * half;
        v2f a, b;
        a.x = h1S[l16][ka];
        a.y = h1S[l16][ka + 1];
        b.x = W2t[(size_t)ka * CC + ncol];
        b.y = W2t[(size_t)(ka + 1) * CC + ncol];
        acc2 = __builtin_amdgcn_wmma_f32_16x16x4_f32(false, a, false, b,
                                                     (short)0, acc2, false, false);
    }
    float bb2 = b2t[ncol];
    #pragma unroll
    for (int r = 0; r < 8; ++r) {
        int rr = r + 8 * half;
        if (rr < ecount)
            atomicAdd(&out[(size_t)cS[rr] * CC + ncol], acc2[r] + bb2);
    }
}

// ---------------------------------------------------------------------------
// h = (residual? h : 0) + LayerNorm(relu(in), g, beta)   (one block per row)
// ---------------------------------------------------------------------------
__global__ __launch_bounds__(128)
void relu_ln(const float* __restrict__ in, const float* __restrict__ g,
             const float* __restrict__ beta, float* __restrict__ h, int residual)
{
    int row = blockIdx.x, c = threadIdx.x;
    float v = fmaxf(in[(size_t)row * CC + c], 0.0f);
    __shared__ float red[CC];
    red[c] = v; __syncthreads();
    for (int s = 64; s > 0; s >>= 1) { if (c < s) red[c] += red[c + s]; __syncthreads(); }
    float mean = red[0] * (1.0f / CC);
    __syncthreads();
    float d = v - mean;
    red[c] = d * d; __syncthreads();
    for (int s = 64; s > 0; s >>= 1) { if (c < s) red[c] += red[c + s]; __syncthreads(); }
    float var = red[0] * (1.0f / CC);
    float nv = d / sqrtf(var + 1e-5f) * g[c] + beta[c];
    size_t o = (size_t)row * CC + c;
    h[o] = residual ? (h[o] + nv) : nv;
}

// ---------------------------------------------------------------------------
// pooling
// ---------------------------------------------------------------------------
__global__ void zero_f(float* p, int n) {
    int i = blockIdx.x * 256 + threadIdx.x;
    if (i < n) p[i] = 0.0f;
}
__global__ __launch_bounds__(128)
void pool_acc(const float* __restrict__ h, const int* __restrict__ batch,
              float* __restrict__ pooled, float* __restrict__ counts)
{
    int i = blockIdx.x, c = threadIdx.x;
    int b = batch[i];
    atomicAdd(&pooled[(size_t)b * CC + c], h[(size_t)i * CC + c]);
    if (c == 0) atomicAdd(&counts[b], 1.0f);
}
__global__ __launch_bounds__(128)
void pool_div(float* pooled, const float* counts)
{
    int b = blockIdx.x, c = threadIdx.x;
    pooled[(size_t)b * CC + c] /= fmaxf(counts[b], 1.0f);
}

// ---------------------------------------------------------------------------
// per-sample 8-head cross attention (H=8, HD=16): tiny, VALU is right here
// ---------------------------------------------------------------------------
__global__ __launch_bounds__(128)
void attn_kernel(const float* __restrict__ Q, const float* __restrict__ Km,
                 const float* __restrict__ V, float* __restrict__ att)
{
    int b = blockIdx.x, tid = threadIdx.x;
    __shared__ float q[CC], k[CC], v[CC], s[8][8], p[8][8];
    q[tid] = Q[(size_t)b * CC + tid];
    k[tid] = Km[(size_t)b * CC + tid];
    v[tid] = V[(size_t)b * CC + tid];
    __syncthreads();
    if (tid < 64) {
        int hh = tid >> 3, gg = tid & 7;
        float a = 0.0f;
        for (int d = 0; d < 16; ++d) a += q[hh * 16 + d] * k[gg * 16 + d];
        s[hh][gg] = a * 0.25f;                       // 1/sqrt(16)
    }
    __syncthreads();
    if (tid < 8) {
        float mx = -1e30f;
        for (int gg = 0; gg < 8; ++gg) mx = fmaxf(mx, s[tid][gg]);
        float sum = 0.0f;
        for (int gg = 0; gg < 8; ++gg) { float e = expf(s[tid][gg] - mx); p[tid][gg] = e; sum += e; }
        float inv = 1.0f / sum;
        for (int gg = 0; gg < 8; ++gg) p[tid][gg] *= inv;
    }
    __syncthreads();
    int hh = tid >> 4, d = tid & 15;
    float a = 0.0f;
    for (int gg = 0; gg < 8; ++gg) a += p[hh][gg] * v[gg * 16 + d];
    att[(size_t)b * CC + tid] = a;
}

// ---------------------------------------------------------------------------
// small plain GEMM for skinny shapes (K=10 or Nc<=4)
// ---------------------------------------------------------------------------
__global__ void gemm_small(const float* __restrict__ A, const float* __restrict__ W,
                           const float* __restrict__ bias, float* __restrict__ out,
                           int M, int Nc, int K, int act)
{
    int i = blockIdx.x * 256 + threadIdx.x;
    if (i >= M * Nc) return;
    int m = i / Nc, n = i % Nc;
    float acc = bias ? bias[n] : 0.0f;
    for (int k = 0; k < K; ++k) acc += A[(size_t)m * K + k] * W[(size_t)k * Nc + n];
    if (act) acc = fmaxf(acc, 0.0f);
    out[i] = acc;
}

// ---------------------------------------------------------------------------
// gru_in = [cross || positional_encoding(ts)]   (192 threads / row)
// ---------------------------------------------------------------------------
__global__ __launch_bounds__(192)
void build_gru_in(const float* __restrict__ cross, const int* __restrict__ ts,
                  float* __restrict__ gin)
{
    int b = blockIdx.x, c = threadIdx.x;
    float v;
    if (c < CC) {
        v = cross[(size_t)b * CC + c];
    } else {
        int j = c - CC;
        float pos = (float)ts[b];
        int i2 = j >> 1;
        float div = expf((float)(2 * i2) * (-logf(10000.0f) / 64.0f));
        v = (j & 1) ? cosf(pos * div) : sinf(pos * div);
    }
    gin[(size_t)b * 192 + c] = v;
}

// GRU step with h_prev = 0 (so gh = bhh)
__global__ __launch_bounds__(128)
void gru_elem(const float* __restrict__ gi, const float* __restrict__ bhh,
              float* __restrict__ hnew)
{
    int b = blockIdx.x, c = threadIdx.x;
    float ir = gi[(size_t)b * 384 + c];
    float iz = gi[(size_t)b * 384 + 128 + c];
    float in = gi[(size_t)b * 384 + 256 + c];
    float hr = bhh[c], hz = bhh[128 + c], hn = bhh[256 + c];
    float r = 1.0f / (1.0f + expf(-(ir + hr)));
    float z = 1.0f / (1.0f + expf(-(iz + hz)));
    float n = tanhf(in + r * hn);
    hnew[(size_t)b * CC + c] = (1.0f - z) * n;   // + z*0
}

__global__ __launch_bounds__(384)
void concat3(const float* __restrict__ a, const float* __restrict__ b_,
             const float* __restrict__ c_, float* __restrict__ out)
{
    int b = blockIdx.x, c = threadIdx.x;
    float v = (c < 128) ? a[(size_t)b * 128 + c]
            : (c < 256) ? b_[(size_t)b * 128 + (c - 128)]
                        : c_[(size_t)b * 128 + (c - 256)];
    out[(size_t)b * 384 + c] = v;
}

__device__ __forceinline__ float softplusf(float x) {
    return x > 20.0f ? x : log1pf(expf(x));
}

__global__ void final_head(const float* __restrict__ y, const float* __restrict__ sel,
                           float* __restrict__ out)
{
    int b = blockIdx.x * 256 + threadIdx.x;
    if (b >= BB) return;
    out[b * 7 + 0] = y[b * 4 + 0];
    out[b * 7 + 1] = softplusf(y[b * 4 + 1]) + 0.01f;
    out[b * 7 + 2] = softplusf(y[b * 4 + 2]) + 1.0f;
    out[b * 7 + 3] = softplusf(y[b * 4 + 3]) + 0.01f;
    out[b * 7 + 4] = sel[b * 3 + 0];
    out[b * 7 + 5] = sel[b * 3 + 1];
    out[b * 7 + 6] = sel[b * 3 + 2];
}

// ---------------------------------------------------------------------------
// host orchestration
// ---------------------------------------------------------------------------
enum {
    P_atom_W = 0, P_atom_b, P_atom_g, P_atom_beta,
    P_cond_W1, P_cond_b1, P_cond_W2, P_cond_b2,
    P_wln_W1, P_wln_b1, P_wln_W2, P_wln_b2, P_wln_Ws, P_wln_bs, P_wln_g, P_wln_beta,
    P_q_W, P_q_b, P_k_W, P_k_b, P_v_W, P_v_b, P_o_W, P_o_b,
    P_gru_Wih, P_gru_Whh, P_gru_bih, P_gru_bhh,
    P_fus_W1, P_fus_b1, P_fus_W2, P_fus_b2,
    P_y_W1, P_y_b1, P_y_W2, P_y_b2,
    P_s_W1, P_s_b1, P_s_W2, P_s_b2, P_COUNT
};

static void encode_graph(const float* x, const int* ei, const int* ea,
                         const float* const* P,
                         float* h, float* tmp, int* ints, int* ids,
                         hipStream_t stream)
{
    const int* row = ei;
    const int* col = ei + EE;

    // bucket edges by type (same set of adds regardless of intra-type order)
    bucket_init<<<1, 32, 0, stream>>>(ints);
    bucket_count<<<(EE + 255) / 256, 256, 0, stream>>>(ea, EE, ints);
    bucket_offsets<<<1, 1, 0, stream>>>(ints);
    bucket_scatter<<<(EE + 255) / 256, 256, 0, stream>>>(ea, EE, ints, ids);

    // atom embedding: h = LN(relu(x @ atom_W + atom_b))
    launch_gemm(x, P[P_atom_W], P[P_atom_b], tmp, NN, CC, 32, 0, 0, stream);
    relu_ln<<<NN, 128, 0, stream>>>(tmp, P[P_atom_g], P[P_atom_beta], h, 0);

    for (int l = 0; l < NLAY; ++l) {
        const float* Ws = P[P_wln_Ws]  + (size_t)l * CC * CC;
        const float* bs = P[P_wln_bs]  + (size_t)l * CC;
        const float* W1 = P[P_wln_W1]  + (size_t)l * NTY * 256 * CC;
        const float* b1 = P[P_wln_b1]  + (size_t)l * NTY * CC;
        const float* W2 = P[P_wln_W2]  + (size_t)l * NTY * CC * CC;
        const float* b2 = P[P_wln_b2]  + (size_t)l * NTY * CC;
        const float* g  = P[P_wln_g]   + (size_t)l * CC;
        const float* bt = P[P_wln_beta]+ (size_t)l * CC;

        // tmp = h @ Ws + bs
        launch_gemm(h, Ws, bs, tmp, NN, CC, CC, 0, 0, stream);
        // tmp += scatter_col( relu([h_row||h_col]@W1_t + b1_t) @ W2_t + b2_t )
        edge_mlp_wmma<<<EE / 16 + NTY, 256, 0, stream>>>(
            h, row, col, ids, ints, W1, b1, W2, b2, tmp);
        // h = h + LN(relu(tmp))
        relu_ln<<<NN, 128, 0, stream>>>(tmp, g, bt, h, 1);
    }
}

extern "C" void kernel_launch(void* const* d_in, const int* in_sizes, int n_in,
                              void* d_out, int out_size, void* d_ws, size_t ws_size,
                              hipStream_t stream)
{
    (void)in_sizes; (void)n_in; (void)out_size; (void)ws_size;

    const float* x_r     = (const float*)d_in[0];
    const int*   ei_r    = (const int*)d_in[1];
    const int*   ea_r    = (const int*)d_in[2];
    const int*   batch_r = (const int*)d_in[3];
    const float* x_p     = (const float*)d_in[4];
    const int*   ei_p    = (const int*)d_in[5];
    const int*   ea_p    = (const int*)d_in[6];
    const int*   batch_p = (const int*)d_in[7];
    const float* cond    = (const float*)d_in[8];
    const int*   ts      = (const int*)d_in[9];

    const float* P[P_COUNT];
    for (int i = 0; i < P_COUNT; ++i) P[i] = (const float*)d_in[10 + i];

    // ---- workspace carve-out (floats) ----
    float* ws = (float*)d_ws;
    size_t off = 0;
    float* h        = ws + off; off += (size_t)NN * CC;   // reused for both graphs
    float* tmp      = ws + off; off += (size_t)NN * CC;
    float* pooled_r = ws + off; off += (size_t)BB * CC;
    float* pooled_p = ws + off; off += (size_t)BB * CC;
    float* counts   = ws + off; off += BB;
    float* Qb       = ws + off; off += (size_t)BB * CC;
    float* Kb       = ws + off; off += (size_t)BB * CC;
    float* Vb       = ws + off; off += (size_t)BB * CC;
    float* attb     = ws + off; off += (size_t)BB * CC;
    float* crossb   = ws + off; off += (size_t)BB * CC;
    float* condh    = ws + off; off += (size_t)BB * CC;
    float* condo    = ws + off; off += (size_t)BB * CC;
    float* gin      = ws + off; off += (size_t)BB * 192;
    float* gi       = ws + off; off += (size_t)BB * 384;
    float* hnew     = ws + off; off += (size_t)BB * CC;
    float* fin      = ws + off; off += (size_t)BB * 384;
    float* fush     = ws + off; off += (size_t)BB * CC;
    float* fused    = ws + off; off += (size_t)BB * CC;
    float* y1       = ws + off; off += (size_t)BB * CC;
    float* y4       = ws + off; off += (size_t)BB * 4;
    float* s1       = ws + off; off += (size_t)BB * 64;
    float* sel3     = ws + off; off += (size_t)BB * 3;
    int*   ints     = (int*)(ws + off); off += 32;
    int*   ids      = (int*)(ws + off);                    // EE ints

    // ---- graph R ----
    encode_graph(x_r, ei_r, ea_r, P, h, tmp, ints, ids, stream);
    zero_f<<<(BB * CC + 255) / 256, 256, 0, stream>>>(pooled_r, BB * CC);
    zero_f<<<(BB + 255) / 256, 256, 0, stream>>>(counts, BB);
    pool_acc<<<NN, 128, 0, stream>>>(h, batch_r, pooled_r, counts);
    pool_div<<<BB, 128, 0, stream>>>(pooled_r, counts);

    // ---- graph P (reuses h/tmp/ids) ----
    encode_graph(x_p, ei_p, ea_p, P, h, tmp, ints, ids, stream);
    zero_f<<<(BB * CC + 255) / 256, 256, 0, stream>>>(pooled_p, BB * CC);
    zero_f<<<(BB + 255) / 256, 256, 0, stream>>>(counts, BB);
    pool_acc<<<NN, 128, 0, stream>>>(h, batch_p, pooled_p, counts);
    pool_div<<<BB, 128, 0, stream>>>(pooled_p, counts);

    // ---- cross attention ----
    launch_gemm(pooled_r, P[P_q_W], P[P_q_b], Qb, BB, CC, CC, 0, 0, stream);
    launch_gemm(pooled_p, P[P_k_W], P[P_k_b], Kb, BB, CC, CC, 0, 0, stream);
    launch_gemm(pooled_p, P[P_v_W], P[P_v_b], Vb, BB, CC, CC, 0, 0, stream);
    attn_kernel<<<BB, 128, 0, stream>>>(Qb, Kb, Vb, attb);
    launch_gemm(attb, P[P_o_W], P[P_o_b], crossb, BB, CC, CC, 0, 0, stream);

    // ---- conditions MLP (K=10 skinny -> plain; then WMMA) ----
    gemm_small<<<(BB * CC + 255) / 256, 256, 0, stream>>>(
        cond, P[P_cond_W1], P[P_cond_b1], condh, BB, CC, 10, 1);
    launch_gemm(condh, P[P_cond_W2], P[P_cond_b2], condo, BB, CC, CC, 0, 0, stream);

    // ---- GRU (h_prev = 0) ----
    build_gru_in<<<BB, 192, 0, stream>>>(crossb, ts, gin);
    launch_gemm(gin, P[P_gru_Wih], P[P_gru_bih], gi, BB, 384, 192, /*transB=*/1, 0, stream);
    gru_elem<<<BB, 128, 0, stream>>>(gi, P[P_gru_bhh], hnew);

    // ---- fusion ----
    concat3<<<BB, 384, 0, stream>>>(crossb, condo, hnew, fin);
    launch_gemm(fin, P[P_fus_W1], P[P_fus_b1], fush, BB, CC, 384, 0, 1, stream);
    launch_gemm(fush, P[P_fus_W2], P[P_fus_b2], fused, BB, CC, CC, 0, 0, stream);

    // ---- heads ----
    launch_gemm(fused, P[P_y_W1], P[P_y_b1], y1, BB, CC, CC, 0, 1, stream);
    gemm_small<<<(BB * 4 + 255) / 256, 256, 0, stream>>>(
        y1, P[P_y_W2], P[P_y_b2], y4, BB, 4, CC, 0);
    launch_gemm(fused, P[P_s_W1], P[P_s_b1], s1, BB, 64, CC, 0, 1, stream);
    gemm_small<<<(BB * 3 + 255) / 256, 256, 0, stream>>>(
        s1, P[P_s_W2], P[P_s_b2], sel3, BB, 3, 64, 0);

    final_head<<<(BB + 255) / 256, 256, 0, stream>>>(y4, sel3, (float*)d_out);
}